// SAGPooling_37744172597492
// MI455X (gfx1250) — compile-verified
//
#include <hip/hip_runtime.h>
#include <hip/hip_bf16.h>
#include <math.h>

#define NB    64
#define NN    2048
#define ND    128
#define NDEG  16
#define KKEEP 1024
#define NTOT  (NB * NN)          // 131072
#define NE    (NB * NN * NDEG)   // 2097152

typedef __attribute__((ext_vector_type(2))) float v2f;
typedef __attribute__((ext_vector_type(8))) float v8f;

// ---- flat float offsets inside d_out (outputs concatenated in return order) ----
#define OFF_XOUT   0
#define OFF_EIDX   (NB * KKEEP * ND)          // 8388608
#define OFF_EATTR  (OFF_EIDX + 2 * NE)        // 12582912
#define OFF_EMASK  (OFF_EATTR + NE)           // 14680064
#define OFF_NBATCH (OFF_EMASK + NE)           // 16777216
#define OFF_PERM   (OFF_NBATCH + NB * KKEEP)  // 16842752
#define OFF_SPERM  (OFF_PERM + NB * KKEEP)    // 16908288

// -------------------------------------------------------------------------
// Phase 1: dual GEMV via f32 WMMA.  Each wave handles 16 node rows.
// A (16x4 f32) = x[rows, k..k+3]; B (4x16 f32) col0 = w_rel, col1 = w_root.
// B is built branch-free: every lane does an unconditional 8-byte load from
// a per-lane-selected weight pointer, then a 0/1 mask zeroes columns >= 2.
// C column 0 -> dot(x, w_rel), column 1 -> dot(x, w_root). 32 WMMAs / wave.
// -------------------------------------------------------------------------
__global__ void __launch_bounds__(256)
gemv_score_kernel(const float* __restrict__ x,
                  const float* __restrict__ w_rel,
                  const float* __restrict__ b_rel,
                  const float* __restrict__ w_root,
                  float* __restrict__ d_rel,
                  float* __restrict__ score,
                  int*   __restrict__ keep,
                  int*   __restrict__ remap)
{
    const int lane  = threadIdx.x & 31;
    const int wave  = threadIdx.x >> 5;
    const int base  = (blockIdx.x * 8 + wave) * 16;   // first of 16 node rows

    const int m     = lane & 15;        // A: row M / B: column N
    const int khalf = (lane >> 4) * 2;  // which k-pair of the 4-wide step

    const float* xrow = x + (size_t)(base + m) * ND + khalf;

    // one pointer cndmask outside the loop; all lanes load valid memory
    const float* wsel = (m == 1) ? w_root : w_rel;
    const float  cmask = (m < 2) ? 1.0f : 0.0f;   // zero B columns 2..15

    v8f c = {};
#pragma unroll
    for (int k = 0; k < ND; k += 4) {
        v2f a = *(const v2f*)(xrow + k);          // global_load_b64
        v2f w = *(const v2f*)(wsel + k + khalf);  // global_load_b64 (L1-hot)
        v2f b;
        b.x = w.x * cmask;
        b.y = w.y * cmask;
        c = __builtin_amdgcn_wmma_f32_16x16x4_f32(false, a, false, b,
                                                  (short)0, c, false, false);
    }

    // C/D layout: VGPR j -> M=j (lanes 0-15, N=lane) or M=j+8 (lanes 16-31, N=lane-16)
    const float brel = b_rel[0];
#pragma unroll
    for (int j = 0; j < 8; ++j) {
        float v = c[j];
        if (lane == 0)       d_rel[base + j]     = v;            // N=0 -> w_rel dots
        else if (lane == 16) d_rel[base + 8 + j] = v;
        else if (lane == 1)  score[base + j]     = v + brel;     // N=1 -> w_root dots
        else if (lane == 17) score[base + 8 + j] = v + brel;
    }
    if (lane < 16) { keep[base + lane] = 0; remap[base + lane] = 0; }
}

// -------------------------------------------------------------------------
// Phase 2: scalar edge scatter  score[dst] += d_rel[src]
// -------------------------------------------------------------------------
__global__ void __launch_bounds__(256)
scatter_kernel(const int* __restrict__ ei,
               const float* __restrict__ d_rel,
               float* __restrict__ score)
{
    int e = blockIdx.x * blockDim.x + threadIdx.x;
    if (e >= NE) return;
    int s = ei[e];
    int d = ei[NE + e];
    unsafeAtomicAdd(&score[d], d_rel[s]);   // -> global_atomic_add_f32
}

// ---- monotone float <-> uint mapping so unsigned compare == float compare ----
__device__ __forceinline__ unsigned int f2ord(float f) {
    unsigned int u = __float_as_uint(f);
    return (u & 0x80000000u) ? ~u : (u | 0x80000000u);
}
__device__ __forceinline__ float ord2f(unsigned int o) {
    unsigned int u = (o & 0x80000000u) ? (o & 0x7FFFFFFFu) : ~o;
    return __uint_as_float(u);
}

// -------------------------------------------------------------------------
// Phase 3: per-graph tanh + top-K via LDS bitonic sort of 2048 packed keys.
// key = (orderable(score) << 32) | (N-1-idx)  => descending sort gives
// descending score with lowest-index tie-break (lax.top_k semantics).
// Emits perm / s_perm / new_batch and marks keep/remap.
// -------------------------------------------------------------------------
__global__ void __launch_bounds__(1024)
topk_kernel(const float* __restrict__ score,
            float* __restrict__ out,
            int*   __restrict__ perm_i,
            float* __restrict__ sval,
            int*   __restrict__ keep,
            int*   __restrict__ remap)
{
    __shared__ unsigned long long keys[NN];   // 16 KB of 320 KB WGP LDS
    const int g = blockIdx.x;
    const int t = threadIdx.x;

    for (int i = t; i < NN; i += 1024) {
        float s = tanhf(score[g * NN + i]);
        keys[i] = ((unsigned long long)f2ord(s) << 32) |
                  (unsigned int)(NN - 1 - i);
    }
    __syncthreads();

    for (int k = 2; k <= NN; k <<= 1) {
        for (int j = k >> 1; j > 0; j >>= 1) {
            for (int i = t; i < NN; i += 1024) {
                int ixj = i ^ j;
                if (ixj > i) {
                    unsigned long long a = keys[i], b = keys[ixj];
                    bool desc = ((i & k) == 0);
                    if (desc ? (a < b) : (a > b)) { keys[i] = b; keys[ixj] = a; }
                }
            }
            __syncthreads();
        }
    }

    // top KKEEP == blockDim: one output per thread, already sorted descending
    unsigned long long kk = keys[t];
    int   idx  = NN - 1 - (int)(unsigned int)(kk & 0xFFFFFFFFu);
    float val  = ord2f((unsigned int)(kk >> 32));
    int   node = g * NN + idx;
    int   r    = g * KKEEP + t;

    out[OFF_PERM   + r] = (float)node;
    out[OFF_SPERM  + r] = val;
    out[OFF_NBATCH + r] = (float)g;
    perm_i[r] = node;
    sval[r]   = val;
    keep[node]  = 1;
    remap[node] = r;
}

// -------------------------------------------------------------------------
// Phase 4: x_out = x[perm] * s_perm   (float4 coalesced gather)
// -------------------------------------------------------------------------
__global__ void __launch_bounds__(256)
gather_kernel(const float* __restrict__ x,
              const int* __restrict__ perm_i,
              const float* __restrict__ sval,
              float* __restrict__ out)
{
    int tid = blockIdx.x * blockDim.x + threadIdx.x;    // B*K*D/4 threads
    int r   = tid >> 5;            // 32 float4 chunks per 128-wide row
    int c   = (tid & 31) << 2;
    int node = perm_i[r];
    float s  = sval[r];
    float4 v = *(const float4*)(x + (size_t)node * ND + c);
    v.x *= s; v.y *= s; v.z *= s; v.w *= s;
    *(float4*)(out + OFF_XOUT + (size_t)r * ND + c) = v;
}

// -------------------------------------------------------------------------
// Phase 5: filter_adj — mask edges with both endpoints kept, remap ids.
// -------------------------------------------------------------------------
__global__ void __launch_bounds__(256)
edge_kernel(const int* __restrict__ ei,
            const float* __restrict__ eattr,
            const int* __restrict__ keep,
            const int* __restrict__ remap,
            float* __restrict__ out)
{
    int e = blockIdx.x * blockDim.x + threadIdx.x;
    if (e >= NE) return;
    int s = ei[e];
    int d = ei[NE + e];
    int m = keep[s] & keep[d];
    out[OFF_EIDX  + e]      = m ? (float)remap[s] : 0.0f;
    out[OFF_EIDX  + NE + e] = m ? (float)remap[d] : 0.0f;
    out[OFF_EATTR + e]      = m ? eattr[e] : 0.0f;
    out[OFF_EMASK + e]      = (float)m;
}

extern "C" void kernel_launch(void* const* d_in, const int* in_sizes, int n_in,
                              void* d_out, int out_size, void* d_ws, size_t ws_size,
                              hipStream_t stream)
{
    const float* x      = (const float*)d_in[0];   // [NTOT, D]
    const int*   ei     = (const int*)  d_in[1];   // [2, NE] flat (src row then dst row)
    const float* eattr  = (const float*)d_in[2];   // [NE]
    // d_in[3] = batch (unused: new_batch derived from graph id)
    const float* w_rel  = (const float*)d_in[4];   // [D]
    const float* b_rel  = (const float*)d_in[5];   // scalar
    const float* w_root = (const float*)d_in[6];   // [D]
    float* out = (float*)d_out;

    // workspace carve (~2.6 MB)
    float* d_rel  = (float*)d_ws;            // [NTOT]
    float* score  = d_rel + NTOT;            // [NTOT]
    int*   keep   = (int*)(score + NTOT);    // [NTOT]
    int*   remap  = keep + NTOT;             // [NTOT]
    int*   perm_i = remap + NTOT;            // [B*K]
    float* sval   = (float*)(perm_i + NB * KKEEP); // [B*K]

    gemv_score_kernel<<<NTOT / 128, 256, 0, stream>>>(x, w_rel, b_rel, w_root,
                                                      d_rel, score, keep, remap);
    scatter_kernel<<<NE / 256, 256, 0, stream>>>(ei, d_rel, score);
    topk_kernel<<<NB, 1024, 0, stream>>>(score, out, perm_i, sval, keep, remap);
    gather_kernel<<<(NB * KKEEP * ND / 4) / 256, 256, 0, stream>>>(x, perm_i, sval, out);
    edge_kernel<<<NE / 256, 256, 0, stream>>>(ei, eattr, keep, remap, out);
}